// GAT_2_Paper_26792005992876
// MI455X (gfx1250) — compile-verified
//
#include <hip/hip_runtime.h>
#include <hip/hip_bf16.h>
#include <math.h>

typedef __attribute__((ext_vector_type(2))) float v2f;
typedef __attribute__((ext_vector_type(8))) float v8f;

#define NEG_SLOPE 0.2f

// ---------------------------------------------------------------------------
// GEMM1: H1[N,8] = X[N,256] @ W1[256,8], via V_WMMA_F32_16X16X4_F32.
// One wave handles a 16-row tile. W1 is staged in LDS pre-padded to 16
// columns (cols 8..15 = 0) so the B-fragment fetch is branch-free and the
// inner loop is global_load_b64 + ds_load_2addr_b32 + v_wmma with full EXEC.
// ---------------------------------------------------------------------------
__global__ __launch_bounds__(256) void gat_gemm1(const float* __restrict__ X,
                                                 const float* __restrict__ W1,
                                                 float* __restrict__ H1,
                                                 int N, int nTiles) {
    __shared__ float w[256 * 16];            // 16 KB padded W1
    for (int i = threadIdx.x; i < 256 * 16; i += 256) {
        int k = i >> 4, col = i & 15;
        w[i] = (col < 8) ? W1[k * 8 + col] : 0.0f;
    }
    __syncthreads();

    int lane = threadIdx.x & 31;
    int tile = blockIdx.x * 8 + (threadIdx.x >> 5);
    if (tile >= nTiles) return;              // whole-wave uniform exit

    int n  = lane & 15;                      // column / row-in-tile index
    int hi = lane >> 4;                      // 0: K pair {0,1}, 1: K pair {2,3}
    int row = tile * 16 + n;
    int rrow = row < N ? row : N - 1;        // clamp reads, keep EXEC all-ones
    const float* xr = X + (size_t)rrow * 256 + 2 * hi;
    const float* wb = w + 2 * hi * 16 + n;   // base of this lane's B column

    v8f c = {};
    #pragma unroll 8
    for (int kk = 0; kk < 256; kk += 4) {
        float2 xv = *(const float2*)(xr + kk);   // A: row rrow, K = kk+2*hi, +1
        v2f a; a.x = xv.x; a.y = xv.y;
        v2f b;                                    // B: K = kk+2*hi(+1), col n
        b.x = wb[kk * 16];
        b.y = wb[kk * 16 + 16];
        c = __builtin_amdgcn_wmma_f32_16x16x4_f32(false, a, false, b,
                                                  (short)0, c, false, false);
    }

    if (n < 8) {
        #pragma unroll
        for (int r = 0; r < 8; ++r) {            // D[M=r+8*hi][n]
            int m = tile * 16 + r + 8 * hi;
            if (m < N) H1[(size_t)m * 8 + n] = c[r];
        }
    }
}

// ---------------------------------------------------------------------------
// GEMM2: H2[N,16] = H[N,8] @ W2[8,16], 2 WMMAs per 16-row tile.
// ---------------------------------------------------------------------------
__global__ __launch_bounds__(256) void gat_gemm2(const float* __restrict__ H,
                                                 const float* __restrict__ W2,
                                                 float* __restrict__ H2,
                                                 int N, int nTiles) {
    __shared__ float w[8 * 16];
    if (threadIdx.x < 128) w[threadIdx.x] = W2[threadIdx.x];
    __syncthreads();

    int lane = threadIdx.x & 31;
    int tile = blockIdx.x * 8 + (threadIdx.x >> 5);
    if (tile >= nTiles) return;

    int n  = lane & 15;
    int hi = lane >> 4;
    int row = tile * 16 + n;
    int rrow = row < N ? row : N - 1;
    const float* hr = H + (size_t)rrow * 8 + 2 * hi;

    v8f c = {};
    #pragma unroll
    for (int kk = 0; kk < 8; kk += 4) {
        v2f a; a.x = hr[kk]; a.y = hr[kk + 1];
        v2f b;
        b.x = w[(kk + 2 * hi) * 16 + n];
        b.y = w[(kk + 2 * hi + 1) * 16 + n];
        c = __builtin_amdgcn_wmma_f32_16x16x4_f32(false, a, false, b,
                                                  (short)0, c, false, false);
    }

    #pragma unroll
    for (int r = 0; r < 8; ++r) {
        int m = tile * 16 + r + 8 * hi;
        if (m < N) H2[(size_t)m * 16 + n] = c[r];
    }
}

// ---------------------------------------------------------------------------
// Node prep: alpha_s/alpha_d = H @ a_src / a_dst, init m=-inf, z=0, zero acc.
// ---------------------------------------------------------------------------
template <int D>
__global__ void gat_node_prep(const float* __restrict__ H,
                              const float* __restrict__ a_src,
                              const float* __restrict__ a_dst,
                              float* __restrict__ aS, float* __restrict__ aD,
                              float* __restrict__ m, float* __restrict__ z,
                              float* __restrict__ acc, int N) {
    int i = blockIdx.x * blockDim.x + threadIdx.x;
    if (i >= N) return;
    const float* h = H + (size_t)i * D;
    float s = 0.0f, d = 0.0f;
    #pragma unroll
    for (int k = 0; k < D; ++k) { s += h[k] * a_src[k]; d += h[k] * a_dst[k]; }
    aS[i] = s; aD[i] = d;
    m[i] = -INFINITY; z[i] = 0.0f;
    float* ai = acc + (size_t)i * D;
    #pragma unroll
    for (int k = 0; k < D; ++k) ai[k] = 0.0f;
}

// ---------------------------------------------------------------------------
// Edge pass 1: per-edge LeakyReLU logit, scatter-max into m[dst].
// ---------------------------------------------------------------------------
__global__ void gat_edge_max(const int* __restrict__ src, const int* __restrict__ dst,
                             const float* __restrict__ aS, const float* __restrict__ aD,
                             float* __restrict__ m, int E) {
    int e = blockIdx.x * blockDim.x + threadIdx.x;
    if (e >= E) return;
    int s = src[e], d = dst[e];
    float x = aS[s] + aD[d];
    x = x > 0.0f ? x : NEG_SLOPE * x;
    atomicMax(m + d, x);    // global_atomic_max_num_f32 (L2)
}

// ---------------------------------------------------------------------------
// Edge pass 2: w = exp(e - m[dst]); z[dst] += w; acc[dst] += H[src]*w.
// ---------------------------------------------------------------------------
template <int D>
__global__ void gat_edge_sum(const int* __restrict__ src, const int* __restrict__ dst,
                             const float* __restrict__ aS, const float* __restrict__ aD,
                             const float* __restrict__ m, const float* __restrict__ H,
                             float* __restrict__ z, float* __restrict__ acc, int E) {
    int e = blockIdx.x * blockDim.x + threadIdx.x;
    if (e >= E) return;
    int s = src[e], d = dst[e];
    float x = aS[s] + aD[d];
    x = x > 0.0f ? x : NEG_SLOPE * x;
    float w = expf(x - m[d]);
    atomicAdd(z + d, w);
    const float* hs = H + (size_t)s * D;
    float* ad = acc + (size_t)d * D;
    #pragma unroll
    for (int k = 0; k < D; ++k) atomicAdd(ad + k, hs[k] * w);
}

// ---------------------------------------------------------------------------
// Layer-1 finish: H = elu(acc / (z+1e-16)) written in-place over H1;
// also re-arms m / z for layer 2.
// ---------------------------------------------------------------------------
__global__ void gat_finish1(const float* __restrict__ acc, float* __restrict__ z,
                            float* __restrict__ H, float* __restrict__ m, int N) {
    int i = blockIdx.x * blockDim.x + threadIdx.x;
    if (i >= N) return;
    float zi = z[i] + 1e-16f;
    #pragma unroll
    for (int k = 0; k < 8; ++k) {
        float v = acc[(size_t)i * 8 + k] / zi;
        H[(size_t)i * 8 + k] = v > 0.0f ? v : expm1f(v);   // elu
    }
    m[i] = -INFINITY;
    z[i] = 0.0f;
}

// ---------------------------------------------------------------------------
// Layer-2 finish: out = acc / (z+1e-16).
// ---------------------------------------------------------------------------
__global__ void gat_finish2(const float* __restrict__ acc, const float* __restrict__ z,
                            float* __restrict__ out, int N) {
    int i = blockIdx.x * blockDim.x + threadIdx.x;
    if (i >= N) return;
    float zi = z[i] + 1e-16f;
    #pragma unroll
    for (int k = 0; k < 16; ++k)
        out[(size_t)i * 16 + k] = acc[(size_t)i * 16 + k] / zi;
}

// ---------------------------------------------------------------------------
extern "C" void kernel_launch(void* const* d_in, const int* in_sizes, int n_in,
                              void* d_out, int out_size, void* d_ws, size_t ws_size,
                              hipStream_t stream) {
    // Inputs (setup_inputs order): V, E, X, W1, a1_src, a1_dst, W2, a2_src, a2_dst
    const int*   Ebuf  = (const int*)  d_in[1];
    const float* X     = (const float*)d_in[2];
    const float* W1    = (const float*)d_in[3];
    const float* a1s   = (const float*)d_in[4];
    const float* a1d   = (const float*)d_in[5];
    const float* W2    = (const float*)d_in[6];
    const float* a2s   = (const float*)d_in[7];
    const float* a2d   = (const float*)d_in[8];
    float*       out   = (float*)d_out;

    const int N = in_sizes[0];          // 100000
    const int E = in_sizes[1] / 2;      // 3200000
    const int* src = Ebuf;
    const int* dst = Ebuf + E;

    // Workspace layout (floats): H1[N*8] H2[N*16] aS[N] aD[N] m[N] z[N] acc[N*16]
    float* w   = (float*)d_ws;
    float* H1  = w;
    float* H2  = H1 + (size_t)N * 8;
    float* aS  = H2 + (size_t)N * 16;
    float* aD  = aS + N;
    float* m   = aD + N;
    float* z   = m + N;
    float* acc = z + N;

    const int nTiles  = (N + 15) / 16;
    const int gBlocks = (nTiles + 7) / 8;
    const int nBlocks = (N + 255) / 256;
    const int eBlocks = (E + 255) / 256;

    // ---- Layer 1 ----
    gat_gemm1<<<gBlocks, 256, 0, stream>>>(X, W1, H1, N, nTiles);
    gat_node_prep<8><<<nBlocks, 256, 0, stream>>>(H1, a1s, a1d, aS, aD, m, z, acc, N);
    gat_edge_max<<<eBlocks, 256, 0, stream>>>(src, dst, aS, aD, m, E);
    gat_edge_sum<8><<<eBlocks, 256, 0, stream>>>(src, dst, aS, aD, m, H1, z, acc, E);
    gat_finish1<<<nBlocks, 256, 0, stream>>>(acc, z, H1, m, N);   // H1 <- elu(norm)

    // ---- Layer 2 ----
    gat_gemm2<<<gBlocks, 256, 0, stream>>>(H1, W2, H2, N, nTiles);
    gat_node_prep<16><<<nBlocks, 256, 0, stream>>>(H2, a2s, a2d, aS, aD, m, z, acc, N);
    gat_edge_max<<<eBlocks, 256, 0, stream>>>(src, dst, aS, aD, m, E);
    gat_edge_sum<16><<<eBlocks, 256, 0, stream>>>(src, dst, aS, aD, m, H2, z, acc, E);
    gat_finish2<<<nBlocks, 256, 0, stream>>>(acc, z, out, N);
}